// EdgeConv_84670985273364
// MI455X (gfx1250) — compile-verified
//
#include <hip/hip_runtime.h>

// Problem sizes (fixed by the reference setup_inputs()).
#define B_ 4
#define C_ 128
#define N_ 8192
#define K_ 16
#define O_ 128          // per-branch channels; output has 2*O_ = 256 channels
#define CH2_ 256

typedef float v2f __attribute__((ext_vector_type(2)));
typedef float v8f __attribute__((ext_vector_type(8)));

// ---------------------------------------------------------------------------
// Kernel 1: dual GEMM  localT[b,n,o] = sum_c W1[o,c]*F[b,c,n]
//                      edgeT [b,n,o] = sum_c W2[o,c]*F[b,c,n]
// 256 threads = 8 waves; wave w owns rows M0=16w..16w+15 of both outputs.
// A fragments (weights) live in registers for the whole kernel.
// ---------------------------------------------------------------------------
#define TPB_TILES 8       // 16-column n-tiles per block -> 128 columns/block
#define FS_STRIDE 132     // padded LDS strides (b64/b128-aligned, conflict-light)
#define TS_STRIDE 132

__global__ __launch_bounds__(256)
void gemm_kernel(const float* __restrict__ feature,
                 const float* __restrict__ W1,
                 const float* __restrict__ W2,
                 float* __restrict__ localT,
                 float* __restrict__ edgeT) {
  __shared__ float Fs [16 * FS_STRIDE];   // feature tile, [n][c]
  __shared__ float TsL[16 * TS_STRIDE];   // transpose staging, [n][o]
  __shared__ float TsE[16 * TS_STRIDE];

  const int b    = blockIdx.y;
  const int tid  = threadIdx.x;
  const int wave = tid >> 5;
  const int lane = tid & 31;
  const int lo   = lane & 15;
  const int hi   = lane >> 4;       // 0: K=0,1  1: K=2,3 (per 16x16x4 f32 layout)
  const int M0   = wave * 16;

  // Preload A fragments for both weight matrices into registers.
  // A 16x4 f32 layout: lanes 0-15 hold row M=lo cols {4k,4k+1}; lanes 16-31 cols {4k+2,4k+3}.
  v2f a1[32], a2[32];
  {
    const float* w1r = W1 + (size_t)(M0 + lo) * C_ + 2 * hi;
    const float* w2r = W2 + (size_t)(M0 + lo) * C_ + 2 * hi;
#pragma unroll
    for (int k = 0; k < 32; ++k) {
      a1[k] = *(const v2f*)(w1r + 4 * k);
      a2[k] = *(const v2f*)(w2r + 4 * k);
    }
  }

  for (int tt = 0; tt < TPB_TILES; ++tt) {
    const int n0 = (blockIdx.x * TPB_TILES + tt) * 16;

    // Cooperative load of feature tile (C_ x 16) into Fs[n][c] (B-fragment order).
    {
      const int c  = tid >> 1;
      const int nb = (tid & 1) * 8;
      const float* src = feature + ((size_t)b * C_ + c) * N_ + n0 + nb;
      float4 f0 = *(const float4*)(src);
      float4 f1 = *(const float4*)(src + 4);
      float v[8] = {f0.x, f0.y, f0.z, f0.w, f1.x, f1.y, f1.z, f1.w};
#pragma unroll
      for (int j = 0; j < 8; ++j) Fs[(nb + j) * FS_STRIDE + c] = v[j];
    }
    __syncthreads();

    v8f accL = {};
    v8f accE = {};
#pragma unroll
    for (int k = 0; k < 32; ++k) {
      // B 4x16 f32: lanes 0-15 hold N=lo rows {4k,4k+1}; lanes 16-31 rows {4k+2,4k+3}.
      v2f bf = *(const v2f*)&Fs[lo * FS_STRIDE + 4 * k + 2 * hi];
      accL = __builtin_amdgcn_wmma_f32_16x16x4_f32(false, a1[k], false, bf,
                                                   (short)0, accL, false, false);
      accE = __builtin_amdgcn_wmma_f32_16x16x4_f32(false, a2[k], false, bf,
                                                   (short)0, accE, false, false);
    }

    // C/D layout: VGPR r, lanes 0-15 -> (M=r, N=lo); lanes 16-31 -> (M=8+r, N=lo).
#pragma unroll
    for (int r = 0; r < 8; ++r) {
      const int o = M0 + r + 8 * hi;
      TsL[lo * TS_STRIDE + o] = accL[r];
      TsE[lo * TS_STRIDE + o] = accE[r];
    }
    __syncthreads();

    // Cooperative transposed store: (b, n, o) with o contiguous (b128 stores).
    {
      const int n = tid >> 4;
      const int j = tid & 15;
      const size_t base = ((size_t)b * N_ + n0 + n) * O_;
      *(float4*)(localT + base + j * 4)      = *(const float4*)&TsL[n * TS_STRIDE + j * 4];
      *(float4*)(localT + base + 64 + j * 4) = *(const float4*)&TsL[n * TS_STRIDE + 64 + j * 4];
      *(float4*)(edgeT  + base + j * 4)      = *(const float4*)&TsE[n * TS_STRIDE + j * 4];
      *(float4*)(edgeT  + base + 64 + j * 4) = *(const float4*)&TsE[n * TS_STRIDE + 64 + j * 4];
    }
    __syncthreads();
  }
}

// ---------------------------------------------------------------------------
// Kernel 2: per-block partial BN statistics (deterministic two-stage reduce).
// 128 threads; thread o owns channel o (central) and channel 128+o (diff).
// ---------------------------------------------------------------------------
#define NCHUNK 64
#define SBLOCKS (B_ * N_ / NCHUNK)    // 512

__global__ __launch_bounds__(128)
void stats_kernel(const float* __restrict__ localT,
                  const float* __restrict__ edgeT,
                  const int*   __restrict__ knn,
                  float* __restrict__ psum,
                  float* __restrict__ psq) {
  const int blk   = blockIdx.x;
  const int b     = blk / (N_ / NCHUNK);
  const int nbase = (blk % (N_ / NCHUNK)) * NCHUNK;
  const int o     = threadIdx.x;

  float sc = 0.f, qc = 0.f, sd = 0.f, qd = 0.f;
  for (int nn = 0; nn < NCHUNK; ++nn) {
    const size_t row = (size_t)b * N_ + nbase + nn;
    const float lc = localT[row * O_ + o];
    sc += lc; qc += lc * lc;
    const int* kn = knn + row * K_;
#pragma unroll
    for (int k = 0; k < K_; ++k) {
      const int i = kn[k];
      const float e = edgeT[((size_t)b * N_ + i) * O_ + o];   // 512B coalesced gather
      const float d = e - lc;
      sd += d; qd += d * d;
    }
  }
  psum[blk * CH2_ + o]        = sc;
  psum[blk * CH2_ + O_ + o]   = sd;
  psq [blk * CH2_ + o]        = qc;
  psq [blk * CH2_ + O_ + o]   = qd;
}

// ---------------------------------------------------------------------------
// Kernel 3: finalize BN -> per-channel scale/shift.
// ---------------------------------------------------------------------------
__global__ __launch_bounds__(256)
void finalize_kernel(const float* __restrict__ psum,
                     const float* __restrict__ psq,
                     const float* __restrict__ gamma,
                     const float* __restrict__ beta,
                     float* __restrict__ scsh) {
  const int c = threadIdx.x;
  float s = 0.f, q = 0.f;
  for (int j = 0; j < SBLOCKS; ++j) {
    s += psum[j * CH2_ + c];
    q += psq [j * CH2_ + c];
  }
  // Central channels were accumulated over B*N (K-fold replication does not
  // change mean/var); diff channels over B*N*K.
  const float cnt  = (c < O_) ? (float)(B_ * N_) : (float)(B_ * N_ * K_);
  const float mean = s / cnt;
  const float var  = q / cnt - mean * mean;
  const float rstd = rsqrtf(var + 1e-5f);
  const float sc   = gamma[c] * rstd;
  scsh[c]        = sc;
  scsh[CH2_ + c] = beta[c] - mean * sc;
}

// ---------------------------------------------------------------------------
// Kernel 4: apply BN + ReLU + mean over K; output (B, 256, N).
// 128 threads; thread o handles channels o and 128+o; LDS transpose so the
// final global stores are contiguous b128 along N.
// ---------------------------------------------------------------------------
#define NT 16
#define OS_STRIDE 20   // pad: 16B-aligned rows, conflict-light

__global__ __launch_bounds__(128)
void apply_kernel(const float* __restrict__ localT,
                  const float* __restrict__ edgeT,
                  const int*   __restrict__ knn,
                  const float* __restrict__ scsh,
                  float* __restrict__ out) {
  __shared__ float outS[CH2_ * OS_STRIDE];

  const int blk = blockIdx.x;
  const int b   = blk / (N_ / NT);
  const int n0  = (blk % (N_ / NT)) * NT;
  const int o   = threadIdx.x;

  const float sc0 = scsh[o],       sh0 = scsh[CH2_ + o];
  const float sc1 = scsh[O_ + o],  sh1 = scsh[CH2_ + O_ + o];

  for (int nn = 0; nn < NT; ++nn) {
    const size_t row = (size_t)b * N_ + n0 + nn;
    const float lc = localT[row * O_ + o];
    const float vc = lc * sc0 + sh0;
    outS[o * OS_STRIDE + nn] = vc > 0.f ? vc : 0.f;

    const int* kn = knn + row * K_;
    float acc = 0.f;
#pragma unroll
    for (int k = 0; k < K_; ++k) {
      const int i = kn[k];
      const float e = edgeT[((size_t)b * N_ + i) * O_ + o];
      const float d = (e - lc) * sc1 + sh1;
      acc += d > 0.f ? d : 0.f;
    }
    outS[(O_ + o) * OS_STRIDE + nn] = acc * (1.0f / K_);
  }
  __syncthreads();

  // 256 channels x 16 n = 1024 float4; 8 per thread, contiguous along n.
#pragma unroll
  for (int rep = 0; rep < 8; ++rep) {
    const int c  = rep * 32 + (o >> 2);
    const int q4 = (o & 3) * 4;
    *(float4*)(out + ((size_t)b * CH2_ + c) * N_ + n0 + q4) =
        *(const float4*)&outS[c * OS_STRIDE + q4];
  }
}

// ---------------------------------------------------------------------------
// Host-side launcher.
// Inputs: 0=feature (B,C,N) f32, 1=knn_inds (B,N,K) i32, 2=W1 (O,C) f32,
//         3=W2 (O,C) f32, 4=gamma (256) f32, 5=beta (256) f32.
// Output: (B, 256, N) f32.
// ---------------------------------------------------------------------------
extern "C" void kernel_launch(void* const* d_in, const int* in_sizes, int n_in,
                              void* d_out, int out_size, void* d_ws, size_t ws_size,
                              hipStream_t stream) {
  const float* feature = (const float*)d_in[0];
  const int*   knn     = (const int*)  d_in[1];
  const float* W1      = (const float*)d_in[2];
  const float* W2      = (const float*)d_in[3];
  const float* gamma   = (const float*)d_in[4];
  const float* beta    = (const float*)d_in[5];
  float*       out     = (float*)d_out;

  // Workspace layout (floats): localT | edgeT | psum | psq | scale+shift
  float* localT = (float*)d_ws;
  float* edgeT  = localT + (size_t)B_ * N_ * O_;
  float* psum   = edgeT  + (size_t)B_ * N_ * O_;
  float* psq    = psum   + (size_t)SBLOCKS * CH2_;
  float* scsh   = psq    + (size_t)SBLOCKS * CH2_;

  gemm_kernel<<<dim3(N_ / (16 * TPB_TILES), B_), 256, 0, stream>>>(
      feature, W1, W2, localT, edgeT);
  stats_kernel<<<SBLOCKS, 128, 0, stream>>>(localT, edgeT, knn, psum, psq);
  finalize_kernel<<<1, 256, 0, stream>>>(psum, psq, gamma, beta, scsh);
  apply_kernel<<<B_ * N_ / NT, 128, 0, stream>>>(localT, edgeT, knn, scsh, out);
}